// SparseLinear_13211319403030
// MI455X (gfx1250) — compile-verified
//
#include <hip/hip_runtime.h>

#define OUT_F 4096
#define IN_F  4096
#define BATCH 256
#define SPMM_CHUNK 1024   // (col,val) pairs staged per block iteration -> 8KB LDS

#ifndef __has_builtin
#define __has_builtin(x) 0
#endif

#define HAVE_ASYNC_LDS __has_builtin(__builtin_amdgcn_global_load_async_to_lds_b32)
#define HAVE_TDM (__has_builtin(__builtin_amdgcn_tensor_load_to_lds) && \
                  __has_builtin(__builtin_amdgcn_s_wait_tensorcnt))

typedef unsigned int u32x4 __attribute__((ext_vector_type(4)));
typedef int          i32x4 __attribute__((ext_vector_type(4)));
typedef int          i32x8 __attribute__((ext_vector_type(8)));

// ---------------------------------------------------------------------------
// CDNA5 async global->LDS staging (ASYNCcnt-tracked DMA into LDS).
// ---------------------------------------------------------------------------
__device__ __forceinline__ void stage_async_b32(const void* g, void* l) {
#if HAVE_ASYNC_LDS
  __builtin_amdgcn_global_load_async_to_lds_b32(
      (__attribute__((address_space(1))) int*)(void*)g,
      (__attribute__((address_space(3))) int*)l,
      /*imm offset*/ 0, /*cpol*/ 0);
#else
  *(volatile int*)l = *(const int*)g;
#endif
}

__device__ __forceinline__ void wait_async0() {
#if __has_builtin(__builtin_amdgcn_s_wait_asynccnt)
  __builtin_amdgcn_s_wait_asynccnt(0);
#else
  asm volatile("s_wait_asynccnt 0" ::: "memory");
#endif
}

#if HAVE_TDM
// ---------------------------------------------------------------------------
// Issue one TDM gather: fetch m (<=16) rows of xT[4096][256] (f32), selected
// by 16-bit indices taken from the LDS-staged CSR cols, into ldsDst
// (contiguous 16x256 f32 tile). D# packing per CDNA5 ISA
// section 8 (groups 0/1) and 8.7 (gather-mode groups 2/3).
// Builtin operands follow the VIMAGE encoding: VADDR0..VADDR3 = D# groups
// 0..3; VADDR4 is unused per ISA -> zero-filled; cpol = 0.
// ---------------------------------------------------------------------------
__device__ __forceinline__ void tdm_gather16(float* ldsDst, const int2* s_pair,
                                             int kbase, int m,
                                             unsigned long long gaddr) {
  unsigned idx[8] = {0u, 0u, 0u, 0u, 0u, 0u, 0u, 0u};
  for (int j = 0; j < m; ++j) {
    unsigned c = (unsigned)s_pair[kbase + j].x & 0xFFFFu;   // col < 4096 -> 16-bit
    idx[j >> 1] |= c << ((j & 1) << 4);
  }
  unsigned lds_off =
      (unsigned)(size_t)(__attribute__((address_space(3))) float*)ldsDst;

  // Force uniform (SGPR) operands for the descriptor groups.
  lds_off = (unsigned)__builtin_amdgcn_readfirstlane((int)lds_off);
  int  mm = __builtin_amdgcn_readfirstlane(m);
  unsigned galo = (unsigned)__builtin_amdgcn_readfirstlane((int)(gaddr & 0xFFFFFFFFull));
  unsigned gahi = (unsigned)__builtin_amdgcn_readfirstlane((int)((gaddr >> 32) & 0xFFFFFFFFull));
#pragma unroll
  for (int q = 0; q < 8; ++q)
    idx[q] = (unsigned)__builtin_amdgcn_readfirstlane((int)idx[q]);

  u32x4 g0;
  g0.x = 1u | (1u << 31);                       // count=1 | gather_mode=1 (idx16)
  g0.y = lds_off;                               // lds_addr
  g0.z = galo;                                  // global_addr[31:0]
  g0.w = (gahi & 0x01FFFFFFu) | (2u << 30);     // global_addr[56:32] | type=2

  i32x8 g1;
  g1[0] = (int)(2u << 16);                      // workgroup_mask=0 | data_size=4B
  g1[1] = (int)(256u << 16);                    // tensor_dim0 = 256 (BATCH)
  g1[2] = (int)(4096u << 16);                   // tensor_dim0.hi=0 | tensor_dim1=4096
  g1[3] = (int)(256u << 16);                    // tensor_dim1.hi=0 | tile_dim0=256
  g1[4] = mm;                                   // tile_dim1 = #valid indices
  g1[5] = 256;                                  // tensor_dim0_stride = 256 elements
  g1[6] = 0;                                    // stride0.hi | stride1.lo (ignored)
  g1[7] = 0;                                    // stride1.hi (ignored in gather)

  i32x4 g2, g3;                                 // 16-bit row indices 0..15
  g2[0] = (int)idx[0]; g2[1] = (int)idx[1]; g2[2] = (int)idx[2]; g2[3] = (int)idx[3];
  g3[0] = (int)idx[4]; g3[1] = (int)idx[5]; g3[2] = (int)idx[6]; g3[3] = (int)idx[7];

  i32x8 v4 = {0, 0, 0, 0, 0, 0, 0, 0};          // VADDR4: unused per ISA -> NULL

  __builtin_amdgcn_tensor_load_to_lds(g0, g1, g2, g3, v4, /*cpol*/ 0);
}
#endif  // HAVE_TDM

// ---------------------------------------------------------------------------
// K1: transpose x[BATCH][IN_F] -> xT[IN_F][BATCH] (batch-column = 1KB line).
// ---------------------------------------------------------------------------
__global__ __launch_bounds__(256) void k_transpose(const float* __restrict__ x,
                                                   float* __restrict__ xT) {
  __shared__ float tile[32][33];
  const int tx = threadIdx.x;
  const int ty = threadIdx.y;
  const int cb = blockIdx.x * 32;
  const int bb = blockIdx.y * 32;
#pragma unroll
  for (int j = ty; j < 32; j += 8)
    tile[j][tx] = x[(bb + j) * IN_F + (cb + tx)];
  __syncthreads();
#pragma unroll
  for (int j = ty; j < 32; j += 8)
    xT[(cb + j) * BATCH + (bb + tx)] = tile[tx][j];
}

// ---------------------------------------------------------------------------
// K2-K5: CSR build (zero, histogram, scan, scatter)
// ---------------------------------------------------------------------------
__global__ void k_zero(int* __restrict__ p, int n) {
  int i = blockIdx.x * blockDim.x + threadIdx.x;
  if (i < n) p[i] = 0;
}

__global__ void k_count(const int* __restrict__ rows, int* __restrict__ cnt, int nnz) {
  int i = blockIdx.x * blockDim.x + threadIdx.x;
  if (i < nnz) atomicAdd(&cnt[rows[i]], 1);
}

__global__ __launch_bounds__(1024) void k_scan(int* __restrict__ cursor,
                                               int* __restrict__ rowOff,
                                               int nnz) {
  __shared__ int s[1024];
  const int tid  = threadIdx.x;
  const int base = tid * 4;
  const int c0 = cursor[base + 0];
  const int c1 = cursor[base + 1];
  const int c2 = cursor[base + 2];
  const int c3 = cursor[base + 3];
  const int mySum = c0 + c1 + c2 + c3;
  s[tid] = mySum;
  __syncthreads();
  for (int off = 1; off < 1024; off <<= 1) {
    int v = (tid >= off) ? s[tid - off] : 0;
    __syncthreads();
    s[tid] += v;
    __syncthreads();
  }
  int excl = s[tid] - mySum;
  const int o0 = excl, o1 = excl + c0, o2 = o1 + c1, o3 = o2 + c2;
  rowOff[base + 0] = o0;  cursor[base + 0] = o0;
  rowOff[base + 1] = o1;  cursor[base + 1] = o1;
  rowOff[base + 2] = o2;  cursor[base + 2] = o2;
  rowOff[base + 3] = o3;  cursor[base + 3] = o3;
  if (tid == 1023) rowOff[OUT_F] = o3 + c3;
  (void)nnz;
}

__global__ void k_scatter(const int* __restrict__ rows, const int* __restrict__ cols,
                          const float* __restrict__ vals, int* __restrict__ cursor,
                          int* __restrict__ colS, float* __restrict__ valS, int nnz) {
  int i = blockIdx.x * blockDim.x + threadIdx.x;
  if (i < nnz) {
    int r = rows[i];
    int p = atomicAdd(&cursor[r], 1);
    colS[p] = cols[i];
    valS[p] = vals[i];
  }
}

// ---------------------------------------------------------------------------
// K6: SpMM. One block (8 wave32) per output row; thread t owns batch t.
//  1) (col,val) CSR slice -> LDS via async DMA (ASYNCcnt).
//  2) x batch-columns -> LDS via double-buffered TDM *gather* descriptors
//     (16 indexed 1KB rows per tensor_load_to_lds, TENSORcnt, 1 in flight).
//  3) hot loop: uniform ds_load pair + ds_load column element + v_fmac_f32.
// ---------------------------------------------------------------------------
__global__ __launch_bounds__(256) void k_spmm(const float* __restrict__ xT,
                                              const int* __restrict__ rowOff,
                                              const int* __restrict__ colS,
                                              const float* __restrict__ valS,
                                              const float* __restrict__ bias,
                                              float* __restrict__ out) {
  __shared__ int2  s_pair[SPMM_CHUNK];     // .x = col, .y = bit-cast val (8KB)
#if HAVE_TDM
  __shared__ float s_x[2][16 * BATCH];     // double-buffered gather tiles (32KB)
#endif
  const int row = blockIdx.x;
  const int t   = threadIdx.x;
  const int beg = rowOff[row];
  const int end = rowOff[row + 1];
  const unsigned long long gaddr = (unsigned long long)(size_t)xT;

  float acc = 0.0f;
  for (int cbase = beg; cbase < end; cbase += SPMM_CHUNK) {
    const int n = min(SPMM_CHUNK, end - cbase);
    // --- stage this chunk's (col,val) pairs into LDS via async DMA ---
    for (int i = t; i < n; i += 256) {
      stage_async_b32(&colS[cbase + i], &s_pair[i].x);
      stage_async_b32(&valS[cbase + i], &s_pair[i].y);
    }
    wait_async0();
    __syncthreads();

#if HAVE_TDM
    const int ng = (n + 15) >> 4;
    if (t == 0)
      tdm_gather16(&s_x[0][0], s_pair, 0, min(16, n), gaddr);
    for (int g = 0; g < ng; ++g) {
      const int kb = g << 4;
      const int m  = min(16, n - kb);
      if (t == 0) {
        if (g + 1 < ng) {   // keep one gather in flight behind compute
          const int kb2 = (g + 1) << 4;
          tdm_gather16(&s_x[(g + 1) & 1][0], s_pair, kb2, min(16, n - kb2), gaddr);
          __builtin_amdgcn_s_wait_tensorcnt(1);
        } else {
          __builtin_amdgcn_s_wait_tensorcnt(0);
        }
      }
      __syncthreads();                      // tile (and pairs) visible to all
      const float* xbuf = &s_x[g & 1][0];
#pragma unroll 4
      for (int j = 0; j < m; ++j) {
        const int2 p = s_pair[kb + j];      // uniform ds_load_b64
        acc = fmaf(__int_as_float(p.y), xbuf[j * BATCH + t], acc);
      }
      __syncthreads();                      // before buffer reuse
    }
#else
    // Fallback: coalesced per-lane global gather of xT columns.
#pragma unroll 4
    for (int k = 0; k < n; ++k) {
      const int2 p = s_pair[k];
      acc = fmaf(__int_as_float(p.y), xT[p.x * BATCH + t], acc);
    }
    __syncthreads();
#endif
  }
  out[t * OUT_F + row] = acc + bias[row];
}

// ---------------------------------------------------------------------------
// Host launcher
// ---------------------------------------------------------------------------
extern "C" void kernel_launch(void* const* d_in, const int* in_sizes, int n_in,
                              void* d_out, int out_size, void* d_ws, size_t ws_size,
                              hipStream_t stream) {
  const float* x    = (const float*)d_in[0];   // [BATCH, IN_F]
  const float* vals = (const float*)d_in[1];   // [nnz]
  const int*   rows = (const int*)  d_in[2];   // [nnz]
  const int*   cols = (const int*)  d_in[3];   // [nnz]
  const float* bias = (const float*)d_in[4];   // [OUT_F]
  const int    nnz  = in_sizes[1];
  float*       out  = (float*)d_out;           // [BATCH, OUT_F]
  (void)n_in; (void)out_size; (void)ws_size;

  char*  ws = (char*)d_ws;
  size_t o  = 0;
  auto carve = [&](size_t bytes) -> void* {
    void* p = ws + o;
    o = (o + bytes + 255) & ~(size_t)255;
    return p;
  };
  float* xT     = (float*)carve(sizeof(float) * IN_F * BATCH);
  int*   rowOff = (int*)  carve(sizeof(int) * (OUT_F + 1));
  int*   cursor = (int*)  carve(sizeof(int) * OUT_F);
  int*   colS   = (int*)  carve(sizeof(int) * (size_t)nnz);
  float* valS   = (float*)carve(sizeof(float) * (size_t)nnz);

  k_transpose<<<dim3(IN_F / 32, BATCH / 32), dim3(32, 8), 0, stream>>>(x, xT);
  k_zero<<<(OUT_F + 255) / 256, 256, 0, stream>>>(cursor, OUT_F);
  k_count<<<(nnz + 255) / 256, 256, 0, stream>>>(rows, cursor, nnz);
  k_scan<<<1, 1024, 0, stream>>>(cursor, rowOff, nnz);
  k_scatter<<<(nnz + 255) / 256, 256, 0, stream>>>(rows, cols, vals, cursor, colS, valS, nnz);
  k_spmm<<<OUT_F, 256, 0, stream>>>(xT, rowOff, colS, valS, bias, out);
}